// GATEncoder_43533788512735
// MI455X (gfx1250) — compile-verified
//
#include <hip/hip_runtime.h>

// ---------------------------------------------------------------------------
// GAT encoder for MI455X (gfx1250), fp32 end-to-end.
// Dense node GEMMs use V_WMMA_F32_16X16X4_F32 (fp32 matrix pipe, wave32).
// Edge softmax/aggregation uses atomic max (ordered-uint trick) + atomic f32
// adds, which stay resident in the 192MB L2 for this working set.
// ---------------------------------------------------------------------------

typedef float v2f __attribute__((ext_vector_type(2)));
typedef float v8f __attribute__((ext_vector_type(8)));

#define NEG_SLOPE 0.2f

__device__ __forceinline__ float leaky(float x) { return x > 0.f ? x : NEG_SLOPE * x; }

// Order-preserving float<->uint encoding so atomicMax(u32) == float max.
__device__ __forceinline__ unsigned f2ord(float f) {
  unsigned u = __float_as_uint(f);
  return (u & 0x80000000u) ? ~u : (u | 0x80000000u);
}
__device__ __forceinline__ float ord2f(unsigned u) {
  u = (u & 0x80000000u) ? (u & 0x7FFFFFFFu) : ~u;
  return __uint_as_float(u);
}

// ---------------------------------------------------------------------------
// Zero-init a contiguous u32 region (ws is poisoned by the harness).
__global__ void zero_init(unsigned* __restrict__ p, long long n) {
  long long i = (long long)blockIdx.x * blockDim.x + threadIdx.x;
  if (i < n) p[i] = 0u;
}

// ---------------------------------------------------------------------------
// C[M x Nc] = A[M x K] @ B[K x Nc], all fp32 row-major.
// One wave (32 lanes) per 16x16 C tile; K-loop in steps of 4 via
// V_WMMA_F32_16X16X4_F32. M, K, Nc all multiples of 16 (true for this net).
// A 16x4 layout: lanes 0-15 row M=lane, v0/v1 = K 0/1; lanes 16-31 = K 2/3.
// B 4x16 layout mirrors C striping: lane%16 = column, half selects K pair.
// C/D: VGPR r, lanes 0-15 -> M=r, lanes 16-31 -> M=r+8, N=lane%16.
__global__ void wmma_gemm_f32(const float* __restrict__ A, const float* __restrict__ B,
                              float* __restrict__ C, int K, int Nc) {
  const int mTile = blockIdx.x * 16;
  const int nTile = blockIdx.y * 16;
  const int lane  = threadIdx.x;     // 0..31, block is exactly one wave
  const int l15   = lane & 15;
  const int hi    = lane >> 4;       // 0 or 1

  v8f acc = {};
  const float* arow = A + (long long)(mTile + l15) * K;
  const float* bcol = B + nTile + l15;

  for (int k = 0; k < K; k += 4) {
    v2f a, b;
    a.x = arow[k + 2 * hi];
    a.y = arow[k + 2 * hi + 1];
    b.x = bcol[(long long)(k + 2 * hi) * Nc];
    b.y = bcol[(long long)(k + 2 * hi + 1) * Nc];
    acc = __builtin_amdgcn_wmma_f32_16x16x4_f32(false, a, false, b,
                                                (short)0, acc, false, false);
  }

  float* crow = C + nTile + l15;
  const int mBase = mTile + hi * 8;
#pragma unroll
  for (int r = 0; r < 8; ++r)
    crow[(long long)(mBase + r) * Nc] = acc[r];
}

// ---------------------------------------------------------------------------
// Layer-1 attention logits: a_src/a_dst[n][h] = dot(h1[n,h,:], att[h,:]), H=2, D=32.
__global__ void attn_scores1(const float* __restrict__ h1,
                             const float* __restrict__ att_src,
                             const float* __restrict__ att_dst,
                             float* __restrict__ asrc, float* __restrict__ adst, int N) {
  int n = blockIdx.x * blockDim.x + threadIdx.x;
  if (n >= N) return;
#pragma unroll
  for (int hd = 0; hd < 2; ++hd) {
    const float* hp = h1 + (long long)n * 64 + hd * 32;
    float s = 0.f, d = 0.f;
#pragma unroll
    for (int i = 0; i < 32; ++i) {
      float v = hp[i];
      s += v * att_src[hd * 32 + i];
      d += v * att_dst[hd * 32 + i];
    }
    asrc[n * 2 + hd] = s;
    adst[n * 2 + hd] = d;
  }
}

// Edge source/dest helper: e < E -> (ei[e], ei[E+e]); else self loop.
__device__ __forceinline__ void edge_sd(const int* __restrict__ ei, long long E,
                                        long long e, int& s, int& d) {
  if (e < E) { s = ei[e]; d = ei[E + e]; } else { s = d = (int)(e - E); }
}

// ---- Layer 1 edge passes ---------------------------------------------------
__global__ void edge_max1(const int* __restrict__ ei, long long E, long long Etot,
                          const float* __restrict__ asrc, const float* __restrict__ adst,
                          unsigned* __restrict__ m) {
  long long e = (long long)blockIdx.x * blockDim.x + threadIdx.x;
  if (e >= Etot) return;
  int s, d; edge_sd(ei, E, e, s, d);
#pragma unroll
  for (int hd = 0; hd < 2; ++hd) {
    float ev = leaky(asrc[s * 2 + hd] + adst[d * 2 + hd]);
    atomicMax(&m[d * 2 + hd], f2ord(ev));
  }
}

__global__ void edge_sum1(const int* __restrict__ ei, long long E, long long Etot,
                          const float* __restrict__ asrc, const float* __restrict__ adst,
                          const unsigned* __restrict__ m, float* __restrict__ den) {
  long long e = (long long)blockIdx.x * blockDim.x + threadIdx.x;
  if (e >= Etot) return;
  int s, d; edge_sd(ei, E, e, s, d);
#pragma unroll
  for (int hd = 0; hd < 2; ++hd) {
    float ev = leaky(asrc[s * 2 + hd] + adst[d * 2 + hd]);
    atomicAdd(&den[d * 2 + hd], __expf(ev - ord2f(m[d * 2 + hd])));
  }
}

// Wave-per-edge message aggregation: lane covers features {lane, lane+32}.
__global__ void edge_msg1(const int* __restrict__ ei, long long E, long long Etot,
                          const float* __restrict__ asrc, const float* __restrict__ adst,
                          const unsigned* __restrict__ m, const float* __restrict__ den,
                          const float* __restrict__ h1, float* __restrict__ out1) {
  long long e = (long long)blockIdx.x * (blockDim.x >> 5) + (threadIdx.x >> 5);
  if (e >= Etot) return;
  int lane = threadIdx.x & 31;
  int s, d; edge_sd(ei, E, e, s, d);
#pragma unroll
  for (int hd = 0; hd < 2; ++hd) {
    float ev    = leaky(asrc[s * 2 + hd] + adst[d * 2 + hd]);
    float alpha = __expf(ev - ord2f(m[d * 2 + hd])) / (den[d * 2 + hd] + 1e-16f);
    int i = hd * 32 + lane;
    atomicAdd(&out1[(long long)d * 64 + i], h1[(long long)s * 64 + i] * alpha);
  }
}

// Bias + ReLU -> hmid (written over the h1 buffer by the caller).
__global__ void relu_bias(const float* __restrict__ acc, const float* __restrict__ b1,
                          float* __restrict__ hmid, long long total) {
  long long i = (long long)blockIdx.x * blockDim.x + threadIdx.x;
  if (i >= total) return;
  float v = acc[i] + b1[i & 63];
  hmid[i] = v > 0.f ? v : 0.f;
}

// ---- Layer 2 (mu + ls fused) ----------------------------------------------
__global__ void attn_scores2(const float* __restrict__ hmu, const float* __restrict__ hls,
                             const float* __restrict__ as_mu, const float* __restrict__ ad_mu,
                             const float* __restrict__ as_ls, const float* __restrict__ ad_ls,
                             float* __restrict__ smu, float* __restrict__ dmu,
                             float* __restrict__ sls, float* __restrict__ dls, int N) {
  int n = blockIdx.x * blockDim.x + threadIdx.x;
  if (n >= N) return;
  float a = 0.f, b = 0.f, c = 0.f, d = 0.f;
#pragma unroll
  for (int i = 0; i < 16; ++i) {
    float vm = hmu[(long long)n * 16 + i];
    float vl = hls[(long long)n * 16 + i];
    a += vm * as_mu[i];
    b += vm * ad_mu[i];
    c += vl * as_ls[i];
    d += vl * ad_ls[i];
  }
  smu[n] = a; dmu[n] = b; sls[n] = c; dls[n] = d;
}

__global__ void edge_max2(const int* __restrict__ ei, long long E, long long Etot,
                          const float* __restrict__ smu, const float* __restrict__ dmu,
                          const float* __restrict__ sls, const float* __restrict__ dls,
                          unsigned* __restrict__ m_mu, unsigned* __restrict__ m_ls) {
  long long e = (long long)blockIdx.x * blockDim.x + threadIdx.x;
  if (e >= Etot) return;
  int s, d; edge_sd(ei, E, e, s, d);
  atomicMax(&m_mu[d], f2ord(leaky(smu[s] + dmu[d])));
  atomicMax(&m_ls[d], f2ord(leaky(sls[s] + dls[d])));
}

__global__ void edge_sum2(const int* __restrict__ ei, long long E, long long Etot,
                          const float* __restrict__ smu, const float* __restrict__ dmu,
                          const float* __restrict__ sls, const float* __restrict__ dls,
                          const unsigned* __restrict__ m_mu, const unsigned* __restrict__ m_ls,
                          float* __restrict__ den_mu, float* __restrict__ den_ls) {
  long long e = (long long)blockIdx.x * blockDim.x + threadIdx.x;
  if (e >= Etot) return;
  int s, d; edge_sd(ei, E, e, s, d);
  atomicAdd(&den_mu[d], __expf(leaky(smu[s] + dmu[d]) - ord2f(m_mu[d])));
  atomicAdd(&den_ls[d], __expf(leaky(sls[s] + dls[d]) - ord2f(m_ls[d])));
}

// Wave-per-edge: lanes 0-15 handle mu features, lanes 16-31 handle ls features.
__global__ void edge_msg2(const int* __restrict__ ei, long long E, long long Etot,
                          const float* __restrict__ smu, const float* __restrict__ dmu,
                          const float* __restrict__ sls, const float* __restrict__ dls,
                          const unsigned* __restrict__ m_mu, const float* __restrict__ den_mu,
                          const unsigned* __restrict__ m_ls, const float* __restrict__ den_ls,
                          const float* __restrict__ hmu, const float* __restrict__ hls,
                          float* __restrict__ out_mu, float* __restrict__ out_ls) {
  long long e = (long long)blockIdx.x * (blockDim.x >> 5) + (threadIdx.x >> 5);
  if (e >= Etot) return;
  int lane = threadIdx.x & 31;
  int s, d; edge_sd(ei, E, e, s, d);
  int f = lane & 15;
  if (lane < 16) {
    float ev = leaky(smu[s] + dmu[d]);
    float al = __expf(ev - ord2f(m_mu[d])) / (den_mu[d] + 1e-16f);
    atomicAdd(&out_mu[(long long)d * 16 + f], hmu[(long long)s * 16 + f] * al);
  } else {
    float ev = leaky(sls[s] + dls[d]);
    float al = __expf(ev - ord2f(m_ls[d])) / (den_ls[d] + 1e-16f);
    atomicAdd(&out_ls[(long long)d * 16 + f], hls[(long long)s * 16 + f] * al);
  }
}

__global__ void finalize(const float* __restrict__ out_mu, const float* __restrict__ out_ls,
                         const float* __restrict__ b_mu, const float* __restrict__ b_ls,
                         float* __restrict__ out, int N) {
  long long i = (long long)blockIdx.x * blockDim.x + threadIdx.x;
  long long total = (long long)N * 16;
  if (i >= total) return;
  int f = (int)(i & 15);
  out[i]         = out_mu[i] + b_mu[f];
  out[total + i] = out_ls[i] + b_ls[f];
}

// ---------------------------------------------------------------------------
extern "C" void kernel_launch(void* const* d_in, const int* in_sizes, int n_in,
                              void* d_out, int out_size, void* d_ws, size_t ws_size,
                              hipStream_t stream) {
  const float* x         = (const float*)d_in[0];
  const int*   ei        = (const int*)d_in[1];     // [2, E] row-major (int32)
  const float* W1        = (const float*)d_in[2];   // [128, 64]
  const float* att_src1  = (const float*)d_in[3];   // [2, 32]
  const float* att_dst1  = (const float*)d_in[4];
  const float* b1        = (const float*)d_in[5];   // [64]
  const float* W_mu      = (const float*)d_in[6];   // [64, 16]
  const float* as_mu     = (const float*)d_in[7];
  const float* ad_mu     = (const float*)d_in[8];
  const float* b_mu      = (const float*)d_in[9];
  const float* W_ls      = (const float*)d_in[10];
  const float* as_ls     = (const float*)d_in[11];
  const float* ad_ls     = (const float*)d_in[12];
  const float* b_ls      = (const float*)d_in[13];

  const int       N    = in_sizes[0] / 128;         // 50000 (multiple of 16)
  const long long E    = in_sizes[1] / 2;           // 800000
  const long long Etot = E + N;                     // + self loops

  // ---- workspace layout (fp32 / u32 elements) ----
  float* ws = (float*)d_ws;
  size_t off = 0;
  float* h1      = ws + off; off += (size_t)N * 64;   // gemm1 out, later reused as hmid
  float* a_src1  = ws + off; off += (size_t)N * 2;
  float* a_dst1  = ws + off; off += (size_t)N * 2;
  float* hmu     = ws + off; off += (size_t)N * 16;
  float* hls     = ws + off; off += (size_t)N * 16;
  float* smu     = ws + off; off += (size_t)N;
  float* dmu     = ws + off; off += (size_t)N;
  float* sls     = ws + off; off += (size_t)N;
  float* dls     = ws + off; off += (size_t)N;
  // zero-init region starts here
  size_t zstart = off;
  unsigned* m1   = (unsigned*)(ws + off); off += (size_t)N * 2;
  float* den1    = ws + off; off += (size_t)N * 2;
  float* out1    = ws + off; off += (size_t)N * 64;
  unsigned* m_mu = (unsigned*)(ws + off); off += (size_t)N;
  float* den_mu  = ws + off; off += (size_t)N;
  unsigned* m_ls = (unsigned*)(ws + off); off += (size_t)N;
  float* den_ls  = ws + off; off += (size_t)N;
  float* out_mu  = ws + off; off += (size_t)N * 16;
  float* out_ls  = ws + off; off += (size_t)N * 16;
  const long long zcount = (long long)(off - zstart);

  const int T = 256;
  const dim3 edgeGrid((unsigned)((Etot + T - 1) / T));
  const dim3 waveGrid((unsigned)((Etot + (T / 32) - 1) / (T / 32)));

  // 0) zero the max/denom/accumulator regions
  zero_init<<<dim3((unsigned)((zcount + T - 1) / T)), T, 0, stream>>>(
      (unsigned*)(ws + zstart), zcount);

  // 1) h1 = x @ W1   (M=N, K=128, Nc=64) via fp32 WMMA
  wmma_gemm_f32<<<dim3(N / 16, 64 / 16), 32, 0, stream>>>(x, W1, h1, 128, 64);

  // 2) layer-1 attention logits
  attn_scores1<<<dim3((N + T - 1) / T), T, 0, stream>>>(h1, att_src1, att_dst1,
                                                        a_src1, a_dst1, N);

  // 3) segment softmax + aggregation (layer 1)
  edge_max1<<<edgeGrid, T, 0, stream>>>(ei, E, Etot, a_src1, a_dst1, m1);
  edge_sum1<<<edgeGrid, T, 0, stream>>>(ei, E, Etot, a_src1, a_dst1, m1, den1);
  edge_msg1<<<waveGrid, T, 0, stream>>>(ei, E, Etot, a_src1, a_dst1, m1, den1, h1, out1);

  // 4) hmid = relu(out1 + b1)  (reuse h1 buffer — h1 is dead after edge_msg1)
  relu_bias<<<dim3((unsigned)(((long long)N * 64 + T - 1) / T)), T, 0, stream>>>(
      out1, b1, h1, (long long)N * 64);

  // 5) layer-2 projections via fp32 WMMA  (M=N, K=64, Nc=16)
  wmma_gemm_f32<<<dim3(N / 16, 1), 32, 0, stream>>>(h1, W_mu, hmu, 64, 16);
  wmma_gemm_f32<<<dim3(N / 16, 1), 32, 0, stream>>>(h1, W_ls, hls, 64, 16);

  // 6) layer-2 attention logits (mu + ls fused)
  attn_scores2<<<dim3((N + T - 1) / T), T, 0, stream>>>(hmu, hls, as_mu, ad_mu,
                                                        as_ls, ad_ls, smu, dmu, sls, dls, N);

  // 7) segment softmax + aggregation (layer 2, mu & ls fused per pass)
  edge_max2<<<edgeGrid, T, 0, stream>>>(ei, E, Etot, smu, dmu, sls, dls, m_mu, m_ls);
  edge_sum2<<<edgeGrid, T, 0, stream>>>(ei, E, Etot, smu, dmu, sls, dls, m_mu, m_ls,
                                        den_mu, den_ls);
  edge_msg2<<<waveGrid, T, 0, stream>>>(ei, E, Etot, smu, dmu, sls, dls,
                                        m_mu, den_mu, m_ls, den_ls, hmu, hls,
                                        out_mu, out_ls);

  // 8) d_out = [mu + b_mu ; ls + b_ls]
  finalize<<<dim3((unsigned)(((long long)N * 16 + T - 1) / T)), T, 0, stream>>>(
      out_mu, out_ls, b_mu, b_ls, (float*)d_out, N);
}